// InstanceLoss2d_52544629900023
// MI455X (gfx1250) — compile-verified
//
#include <hip/hip_runtime.h>
#include <hip/hip_bf16.h>

#define Bn 16
#define Cn 4
#define Hn 512
#define Wn 512
#define NPLANE (Bn * Cn)       // 64
#define NSTRIP 4
#define SROWS  (Hn / NSTRIP)   // 128 rows per strip
#define NTHREADS 256
#define NWAVES (NTHREADS / 32)
#define BT 16                  // rows batched in LDS per reduction round
#define NB (SROWS / BT)        // 8 batches per strip

// sigmoid(-5): value of sig(|0-0|) at fully padded positions
#define SIG0 0.00669285092428486f

typedef __attribute__((ext_vector_type(2))) float v2f;
typedef __attribute__((ext_vector_type(8))) float v8f;

// sigmoid(t) with a single transcendental (gfx1250 V_TANH_F32)
__device__ __forceinline__ float fast_sigmoid(float t) {
#if __has_builtin(__builtin_amdgcn_tanhf)
    return fmaf(0.5f, __builtin_amdgcn_tanhf(0.5f * t), 0.5f);
#else
    float e = __builtin_amdgcn_exp2f(t * -1.44269504088896340736f);
    return __builtin_amdgcn_rcpf(1.0f + e);
#endif
}
// sigmoid((x - 0.5) * 10)
__device__ __forceinline__ float sig10(float x) {
    return fast_sigmoid(fmaf(10.0f, x, -5.0f));
}

__device__ __forceinline__ float wave_sum(float v) {
#pragma unroll
    for (int m = 1; m < 32; m <<= 1) v += __shfl_xor(v, m, 32);
    return v;
}

// Phase 1: per (tensor, plane, row-strip):
//   sxPart[tensor][plane][strip][k] = sum_{i in strip} sig(|xs[i+1,k]-xs[i-1,k]|)
//   sy[tensor][plane][i]            = sum_j sig(|xs[i-1,j+1]-xs[i-1,j-1]|)
// Rows are batched BT at a time into LDS; horizontal reductions are done
// per-wave afterwards (2 barriers per BT rows instead of 2 per row).
__global__ __launch_bounds__(NTHREADS) void ph1_structs(
        const float* __restrict__ inp, const float* __restrict__ tgt,
        float* __restrict__ sxPart, float* __restrict__ syOut) {
    const int strip  = blockIdx.x;
    const int plane  = blockIdx.y;
    const int tensor = blockIdx.z;
    const float* __restrict__ src =
        (tensor ? tgt : inp) + (size_t)plane * Hn * Wn;
    const int t  = threadIdx.x;
    const int j0 = 2 * t, j1 = 2 * t + 1;
    const int wv = t >> 5, ln = t & 31;

    __shared__ float sg[BT][Wn + 2];   // sigmoid rows, zero column halos
    if (t < BT) { sg[t][0] = 0.0f; sg[t][Wn + 1] = 0.0f; }

    const int i0 = strip * SROWS;
    float p2a = 0.0f, p2b = 0.0f, p1a = 0.0f, p1b = 0.0f;  // rows r-2, r-1
    float sxa = 0.0f, sxb = 0.0f;
    float* syBase = syOut + ((size_t)tensor * NPLANE + plane) * Hn;

    for (int k = 0; k < NB; ++k) {
        const int rbase = i0 - 1 + BT * k;   // load rows rbase .. rbase+BT-1
        // ---- phase A: load + sigmoid + vertical accumulation (no barriers)
#pragma unroll
        for (int rr = 0; rr < BT; ++rr) {
            const int r = rbase + rr;        // always < Hn inside batches
            float ca = 0.0f, cb = 0.0f;
            if (r >= 0) {
                const float2 v = ((const float2*)(src + (size_t)r * Wn))[t];
                ca = sig10(v.x);
                cb = sig10(v.y);
            }
            if (r >= i0 + 1) {               // vertical bound for i = r-1
                sxa += sig10(fabsf(ca - p2a));
                sxb += sig10(fabsf(cb - p2b));
            }
            sg[rr][j0 + 1] = ca;
            sg[rr][j1 + 1] = cb;
            p2a = p1a; p2b = p1b; p1a = ca; p1b = cb;
        }
        __syncthreads();
        // ---- phase B: horizontal bounds + per-row sums (2 rows per wave)
#pragma unroll
        for (int q = 0; q < BT / NWAVES; ++q) {
            const int rr = wv * (BT / NWAVES) + q;
            const float* row = &sg[rr][0];
            float acc = 0.0f;
#pragma unroll
            for (int m = 0; m < Wn / 32; ++m) {
                const int j = ln + 32 * m;   // conflict-free strided columns
                acc += sig10(fabsf(row[j + 2] - row[j]));
            }
            acc = wave_sum(acc);
            if (ln == 0) syBase[rbase + rr + 1] = acc;  // sy index = r+1
        }
        __syncthreads();                     // sg reused next batch
    }
    // leftover rows r = i0+SROWS-1, i0+SROWS feed only the vertical sums
    for (int r = i0 + SROWS - 1; r <= i0 + SROWS; ++r) {
        float ca = 0.0f, cb = 0.0f;
        if (r < Hn) {
            const float2 v = ((const float2*)(src + (size_t)r * Wn))[t];
            ca = sig10(v.x);
            cb = sig10(v.y);
        }
        sxa += sig10(fabsf(ca - p2a));
        sxb += sig10(fabsf(cb - p2b));
        p2a = p1a; p2b = p1b; p1a = ca; p1b = cb;
    }
    float2 sx2; sx2.x = sxa; sx2.y = sxb;
    ((float2*)(sxPart +
        (((size_t)tensor * NPLANE + plane) * NSTRIP + strip) * (size_t)Wn))[t] = sx2;
}

// Phase 2: per plane: build full sx/sy vectors, sobel([-1,0,1]) + |.| + sum,
// combine input vs target with weight -> planeLoss[plane]
__global__ __launch_bounds__(NTHREADS) void ph2_sobel(
        const float* __restrict__ sxPart, const float* __restrict__ syIn,
        const float* __restrict__ weight, float* __restrict__ planeLoss) {
    const int plane = blockIdx.x;
    const int t  = threadIdx.x;
    const int j0 = 2 * t, j1 = 2 * t + 1;

    __shared__ float s[Wn + 2];
    __shared__ float wred[NWAVES];
    __shared__ float res[4];
    if (t == 0) { s[0] = 0.0f; s[Wn + 1] = 0.0f; }

#pragma unroll
    for (int q = 0; q < 4; ++q) {
        const int tensor = q & 1;     // 0=input, 1=target
        const int isY    = q >> 1;    // 0=sx, 1=sy
        float va, vb;
        if (!isY) {
            const float* base =
                sxPart + (((size_t)tensor * NPLANE + plane) * NSTRIP) * (size_t)Wn;
            va = (j0 == 0) ? (float)Hn * SIG0 : 0.0f;  // sx[0]: all-padding column
            vb = 0.0f;
            if (j0 > 0) {
#pragma unroll
                for (int st = 0; st < NSTRIP; ++st) va += base[(size_t)st * Wn + (j0 - 1)];
            }
#pragma unroll
            for (int st = 0; st < NSTRIP; ++st) vb += base[(size_t)st * Wn + (j1 - 1)];
        } else {
            const float* base = syIn + ((size_t)tensor * NPLANE + plane) * Hn;
            va = base[j0];
            vb = base[j1];
        }
        __syncthreads();           // previous pass done reading s / wred
        s[j0 + 1] = va;
        s[j1 + 1] = vb;
        __syncthreads();
        float part = fabsf(s[j0 + 2] - s[j0]) + fabsf(s[j1 + 2] - s[j1]);
        part = wave_sum(part);
        if ((t & 31) == 0) wred[t >> 5] = part;
        __syncthreads();
        if (t == 0) {
            float acc = 0.0f;
#pragma unroll
            for (int w = 0; w < NWAVES; ++w) acc += wred[w];
            res[q] = acc;
        }
    }
    __syncthreads();
    if (t == 0) {
        const int c = plane % Cn;
        const float w = weight[c];
        const float Txi = res[0], Txt = res[1], Tyi = res[2], Tyt = res[3];
        // each struct scalar = T/4; loss = (|dx| + |dy|)/2  -> factor 1/8
        planeLoss[plane] =
            (fabsf(Txi * w - Txt * w) + fabsf(Tyi * w - Tyt * w)) * 0.125f;
    }
}

// Phase 3: 64 -> 1 reduction as D = ones(16x4) x B(4x16) via V_WMMA_F32_16X16X4_F32.
// B VGPR layout (ISA 7.12.2): VGPR0 = K0 (lanes 0-15) / K2 (lanes 16-31),
// VGPR1 = K1 / K3. D rows are identical column sums; 4 shuffles finish.
__global__ __launch_bounds__(32) void ph3_final(
        const float* __restrict__ planeLoss, float* __restrict__ out) {
    const int lane = threadIdx.x;
    const int n  = lane & 15;
    const int hi = lane >> 4;
#if __has_builtin(__builtin_amdgcn_wmma_f32_16x16x4_f32)
    v2f a; a.x = 1.0f; a.y = 1.0f;                // A = ones(16x4)
    v2f b;
    b.x = planeLoss[(hi * 2 + 0) * 16 + n];       // B[k][n] = loss[k*16+n]
    b.y = planeLoss[(hi * 2 + 1) * 16 + n];
    v8f c = {};
    c = __builtin_amdgcn_wmma_f32_16x16x4_f32(
        /*neg_a=*/false, a, /*neg_b=*/false, b,
        /*c_mod=*/(short)0, c, /*reuse_a=*/false, /*reuse_b=*/false);
    float v = c[0];                               // colsum[n] in every 16-lane half
#pragma unroll
    for (int m = 1; m < 16; m <<= 1) v += __shfl_xor(v, m, 32);
    if (lane == 0) out[0] = v * (1.0f / (float)NPLANE);   // mean over [B,C]
#else
    float v = planeLoss[lane] + planeLoss[lane + 32];
    v = wave_sum(v);
    if (lane == 0) out[0] = v * (1.0f / (float)NPLANE);
#endif
}

extern "C" void kernel_launch(void* const* d_in, const int* in_sizes, int n_in,
                              void* d_out, int out_size, void* d_ws, size_t ws_size,
                              hipStream_t stream) {
    const float* inp = (const float*)d_in[0];
    const float* tgt = (const float*)d_in[1];
    const float* wgt = (const float*)d_in[2];

    float* ws     = (float*)d_ws;
    float* sxPart = ws;                                         // 2*64*4*512 f
    float* sy     = sxPart + (size_t)2 * NPLANE * NSTRIP * Wn;  // 2*64*512 f
    float* pl     = sy + (size_t)2 * NPLANE * Hn;               // 64 f

    dim3 g1(NSTRIP, NPLANE, 2);
    ph1_structs<<<g1, NTHREADS, 0, stream>>>(inp, tgt, sxPart, sy);
    ph2_sobel<<<NPLANE, NTHREADS, 0, stream>>>(sxPart, sy, wgt, pl);
    ph3_final<<<1, 32, 0, stream>>>(pl, (float*)d_out);
}